// PeriodicPadding3d_24592982737350
// MI455X (gfx1250) — compile-verified
//
#include <hip/hip_runtime.h>

// Periodic (wrap) 3D padding: [8,4,128,128,128] f32 -> [8,4,132,132,132] f32.
// Pure bandwidth problem (~0.54 GB traffic -> ~24us floor @ 23.3 TB/s).
//
// Mapping (no div/mod, no guards, no branches):
//   grid  = (6, 33, 32):  blockIdx.z = bc (batch*chan), blockIdx.y = zo0,
//                         blockIdx.x = chunk of 22 output rows in the plane
//   block = (33, 22):     threadIdx.x = float4 slot i in row (132 floats =
//                         33 float4s, 528B pitch => rows 16B aligned),
//                         threadIdx.y = row within chunk (6*22 = 132 exact)
//   Each thread handles 4 z-planes: zo = zo0 + {0,33,66,99} (4*33 = 132),
//   reusing all per-lane index math (only the uniform plane base changes).
// Per thread: 8x 8B-aligned b64 gathers + 4x 16B non-temporal stores = 128B.

#define PAD   2
#define DIN   128
#define DOUT  132
#define ZSTEP 33          // DOUT / 4 planes per thread
#define PLANE_IN  (DIN * DIN)        // 16384 floats
#define PLANE_OUT (DOUT * 33)        // 4356 float4s

typedef float v4f __attribute__((ext_vector_type(4)));
typedef float v2f __attribute__((ext_vector_type(2)));

__global__ __launch_bounds__(726) void periodic_pad3d_kernel(
    const float* __restrict__ in, float* __restrict__ out)
{
    const int i   = threadIdx.x;                   // 0..32 : float4 slot
    const int ty  = threadIdx.y;                   // 0..21 : row in chunk
    const int yo  = blockIdx.x * 22 + ty;          // 0..131
    const int zo0 = blockIdx.y;                    // 0..32  (uniform)
    const int bc  = blockIdx.z;                    // 0..31  (uniform)

    // wrap: src coord = (o - PAD) mod 128
    int y = yo - PAD;  y += (y < 0) ? DIN : 0;  y -= (y >= DIN) ? DIN : 0;

    // Branchless x gather: output floats xo = 4i..4i+3 need x = 4i-2..4i+1
    // (mod 128). xa = 4i-2 wraps only for i==0 (-2 -> 126); for i==32,
    // 4i-2 = 126 is already the correct halo address. xb = 4i wraps only
    // for i==32 (128 -> 0).
    int xa = 4 * i - 2;  xa += (xa < 0) ? DIN : 0;
    int xb = 4 * i;      xb -= (xb >= DIN) ? DIN : 0;

    const int rowoff = y * DIN;                    // per-lane, reused 4x
    const int lin    = ty * 33 + i;                // output slot, reused 4x

    const float* base_in = in + (size_t)bc * DIN * PLANE_IN;
    v4f* base_out = (v4f*)out
                  + (size_t)bc * DOUT * PLANE_OUT
                  + (size_t)(blockIdx.x * 22) * 33;

#pragma unroll
    for (int k = 0; k < 4; ++k) {
        const int zo = zo0 + k * ZSTEP;            // uniform per block
        int z = zo - PAD;  z += (z < 0) ? DIN : 0;  z -= (z >= DIN) ? DIN : 0;

        const float* plane = base_in + (size_t)z * PLANE_IN;
        v2f a = *(const v2f*)(plane + rowoff + xa);   // 8B aligned
        v2f b = *(const v2f*)(plane + rowoff + xb);

        v4f v = { a.x, a.y, b.x, b.y };
        // Write-once output, never re-read -> non-temporal keeps L2 for
        // the input halo re-reads.
        __builtin_nontemporal_store(v, base_out + (size_t)zo * PLANE_OUT + lin);
    }
}

extern "C" void kernel_launch(void* const* d_in, const int* in_sizes, int n_in,
                              void* d_out, int out_size, void* d_ws, size_t ws_size,
                              hipStream_t stream) {
    (void)in_sizes; (void)n_in; (void)d_ws; (void)ws_size; (void)out_size;
    const float* x = (const float*)d_in[0];
    float* out = (float*)d_out;

    dim3 grid(6, ZSTEP, 32);    // 6 row-chunks * 33 zo0 * (8*4) bc
    dim3 block(33, 22);         // 726 threads = 23 wave32s, 1.4% lane waste

    hipLaunchKernelGGL(periodic_pad3d_kernel, grid, block, 0, stream, x, out);
}